// TourConstructor_59700045414695
// MI455X (gfx1250) — compile-verified
//
#include <hip/hip_runtime.h>
#include <float.h>
#include <stdint.h>

// Greedy hard-permutation (straight-through) for B=256 batches of 256x256 f32.
// One workgroup per batch; entire score tile resident in CDNA5's 320KB LDS,
// staged with gfx1250 async global->LDS loads. Incremental per-row argmax
// caching makes each of the 256 sequential steps ~O(N) instead of O(N^2).

#define NN 256

typedef __attribute__((ext_vector_type(4))) float v4f;

// Dynamic LDS layout (bytes)
#define OFF_S       0                       // float s[256*256]        (262144 B)
#define OFF_ROWVAL  (NN * NN * 4)           // float rowVal[256]
#define OFF_ROWCOL  (OFF_ROWVAL + NN * 4)   // int   rowCol[256]
#define OFF_COLMSK  (OFF_ROWCOL + NN * 4)   // int   colMask[256]
#define OFF_ASGCOL  (OFF_COLMSK + NN * 4)   // int   asgCol[256]
#define OFF_REDV    (OFF_ASGCOL + NN * 4)   // float redV[8]
#define OFF_REDI    (OFF_REDV + 32)         // int   redI[8]
#define OFF_RC      (OFF_REDI + 32)         // int   s_rc[2]
#define SMEM_BYTES  ((OFF_RC + 8 + 63) & ~63)

__device__ __forceinline__ uint32_t lds_off_u32(const void* p) {
    // Flat pointers into LDS carry the group offset in their low 32 bits.
    return (uint32_t)(uintptr_t)p;
}

__global__ void __launch_bounds__(256)
greedy_perm_kernel(const float* __restrict__ soft, float* __restrict__ out) {
    extern __shared__ char smem[];
    float* s       = (float*)(smem + OFF_S);
    float* rowVal  = (float*)(smem + OFF_ROWVAL);
    int*   rowCol  = (int*)  (smem + OFF_ROWCOL);
    int*   colMask = (int*)  (smem + OFF_COLMSK);
    int*   asgCol  = (int*)  (smem + OFF_ASGCOL);
    float* redV    = (float*)(smem + OFF_REDV);
    int*   redI    = (int*)  (smem + OFF_REDI);
    int*   s_rc    = (int*)  (smem + OFF_RC);

    const int t = threadIdx.x;                 // 0..255, 8 wave32s
    const int b = blockIdx.x;                  // batch
    const float* gsrc = soft + (size_t)b * (NN * NN);
    float*       gdst = out  + (size_t)b * (NN * NN);

    // ---- Phase 1: async DMA the 256KB score tile into LDS (CDNA5 path) ----
    // ISA: the instruction offset is added to both the LDS and the global address,
    // i.e. the 24-bit immediate offsets BOTH sides -> one base address pair
    // per thread, 64 bare async-load issues with stride-4096 immediates.
    {
        const uint32_t loff0 = lds_off_u32(s) + (uint32_t)t * 16u;
        const float*   g0    = gsrc + (size_t)t * 4;   // thread's 16B chunk 0
        #pragma unroll
        for (int i = 0; i < 64; ++i) {
            asm volatile("global_load_async_to_lds_b128 %0, %1, off offset:%2"
                         :: "v"(loff0), "v"(g0), "n"(i * 4096)
                         : "memory");
        }
        asm volatile("s_wait_asynccnt 0x0" ::: "memory");
    }
    colMask[t] = 0;
    asgCol[t]  = 0;
    __syncthreads();

    // ---- Phase 2: initial per-row max (first-occurrence argmax) ----
    {
        const float* row = s + t * NN;
        float bv = -FLT_MAX;
        int   bc = 0;
        for (int j = 0; j < NN; ++j) {
            const int c = (j + t) & (NN - 1);          // skew: avoid bank conflicts
            const float v = row[c];
            if (v > bv || (v == bv && c < bc)) { bv = v; bc = c; }
        }
        rowVal[t] = bv;
        rowCol[t] = bc;
    }
    __syncthreads();

    // ---- Phase 3: 256 sequential greedy steps ----
    for (int step = 0; step < NN; ++step) {
        // argmax over rows: prefer larger value, then smaller row (jnp.argmax order)
        float v = rowVal[t];
        int   r = t;
        #pragma unroll
        for (int off = 16; off > 0; off >>= 1) {
            const float ov = __shfl_xor(v, off, 32);
            const int   orr = __shfl_xor(r, off, 32);
            if (ov > v || (ov == v && orr < r)) { v = ov; r = orr; }
        }
        if ((t & 31) == 0) { redV[t >> 5] = v; redI[t >> 5] = r; }
        __syncthreads();
        if (t == 0) {
            float bv = redV[0];
            int   br = redI[0];
            #pragma unroll
            for (int w = 1; w < 8; ++w) {
                if (redV[w] > bv || (redV[w] == bv && redI[w] < br)) {
                    bv = redV[w]; br = redI[w];
                }
            }
            const int c = rowCol[br];
            s_rc[0] = br;
            s_rc[1] = c;
            asgCol[br]  = c;
            colMask[c]  = 1;
            rowVal[br]  = -FLT_MAX;   // row consumed
        }
        __syncthreads();
        const int c = s_rc[1];
        // Rows whose cached argmax column was just consumed must rescan.
        // (Cache stays valid otherwise: removing a non-argmax column can't
        // raise the max, and first-occurrence ties are preserved.)
        if (rowVal[t] != -FLT_MAX && rowCol[t] == c) {
            const float* row = s + t * NN;
            float bv = -FLT_MAX;
            int   bc = 0;
            for (int j = 0; j < NN; ++j) {
                const int cc = (j + t) & (NN - 1);
                if (!colMask[cc]) {
                    const float vv = row[cc];
                    if (vv > bv || (vv == bv && cc < bc)) { bv = vv; bc = cc; }
                }
            }
            rowVal[t] = bv;
            rowCol[t] = bc;
        }
        __syncthreads();
    }

    // ---- Phase 4: stream output once, coalesced non-temporal 16B stores ----
    // out[r][j] = (j == asgCol[r]) ? (1+s)-s : 0   (exact STE arithmetic)
    #pragma unroll 2
    for (int i = 0; i < 64; ++i) {
        const int k   = i * 256 + t;          // 16B chunk; consecutive t -> coalesced
        const int row = k >> 6;               // 64 chunks per 1KB row
        const int c0  = (k & 63) * 4;
        const int ac  = asgCol[row];
        const v4f sv  = *(const v4f*)(s + row * NN + c0);   // ds_load_b128
        v4f o;
        o.x = (c0 + 0 == ac) ? ((1.0f + sv.x) - sv.x) : 0.0f;
        o.y = (c0 + 1 == ac) ? ((1.0f + sv.y) - sv.y) : 0.0f;
        o.z = (c0 + 2 == ac) ? ((1.0f + sv.z) - sv.z) : 0.0f;
        o.w = (c0 + 3 == ac) ? ((1.0f + sv.w) - sv.w) : 0.0f;
        __builtin_nontemporal_store(o, (v4f*)(gdst + (size_t)k * 4));
    }
}

extern "C" void kernel_launch(void* const* d_in, const int* in_sizes, int n_in,
                              void* d_out, int out_size, void* d_ws, size_t ws_size,
                              hipStream_t stream) {
    (void)in_sizes; (void)n_in; (void)out_size; (void)d_ws; (void)ws_size;
    const float* soft = (const float*)d_in[0];
    float*       out  = (float*)d_out;
    hipLaunchKernelGGL(greedy_perm_kernel, dim3(256), dim3(256), SMEM_BYTES, stream,
                       soft, out);
}